// BilateralFilter_41678362640719
// MI455X (gfx1250) — compile-verified
//
#include <hip/hip_runtime.h>
#include <stdint.h>

// Bilateral filter, fused single pass per 16x16 tile.
// x: (4,32,256,256) f32, params: (4,3) f32 -> out: (4,32,256,256) f32.
//
// MI455X reasoning: bandwidth-bound (~90MB moved, ~4us at 23.3TB/s), so we
// read x exactly once per tile via CDNA5 async global->LDS DMA, compute the
// channel mean with chained f32 WMMAs (exact f32 math), and keep everything
// resident in the 320KB/WGP LDS.

typedef float v2f __attribute__((ext_vector_type(2)));
typedef float v8f __attribute__((ext_vector_type(8)));

#define BATCH 4
#define CHANS 32
#define IMG_H 256
#define IMG_W 256
#define TILE 16
#define HALO 2
#define TW 20              // TILE + 2*HALO
#define TELEMS 400         // TW*TW floats per channel tile
#define TPAIRS 200         // b64 pairs per channel tile

// ---- CDNA5 async global->LDS DMA (ASYNCcnt path) ----
__device__ __forceinline__ void async_ld_b64(void* lds, const void* g) {
  uint32_t l = (uint32_t)(uintptr_t)lds;  // flat shared ptr low 32 bits == LDS address
  asm volatile("global_load_async_to_lds_b64 %0, %1, off"
               :: "v"(l), "v"(g) : "memory");
}
__device__ __forceinline__ void wait_async0() {
  asm volatile("s_wait_asynccnt 0x0" ::: "memory");
}

// ---- per-sample scalar prep: sk[25] (masked+normalized), 1/(2*sigma2^2), zero word ----
__global__ void bilateral_prep(const float* __restrict__ params, float* __restrict__ ws) {
  int b = threadIdx.x;
  if (b < BATCH) {
    float p0 = params[b*3+0], p1 = params[b*3+1], p2 = params[b*3+2];
    float sg0 = 1.f / (1.f + __expf(-p0));
    float is5 = (1.f + 2.f*sg0 >= 2.f) ? 1.f : 0.f;
    float s1 = 3.5f + 5.5f * (1.f / (1.f + __expf(-p1)));
    float s2 = 5.5f + 7.5f * (1.f / (1.f + __expf(-p2)));
    float inv2s1 = 1.f / (2.f * s1 * s1);
    float sk[25];
    float sum = 0.f;
    for (int i = 0; i < 5; ++i)
      for (int j = 0; j < 5; ++j) {
        float ci = (float)(i - 2), cj = (float)(j - 2);
        float grid = ci*ci + cj*cj;
        float m = (is5 > 0.5f) ? 1.f
                               : ((fabsf(ci) <= 1.f && fabsf(cj) <= 1.f) ? 1.f : 0.f);
        float v = __expf(-grid * inv2s1) * m;
        sk[i*5+j] = v;
        sum += v;
      }
    float inv = 1.f / sum;
    for (int t = 0; t < 25; ++t) ws[b*32 + t] = sk[t] * inv;
    ws[b*32 + 25] = 1.f / (2.f * s2 * s2);
  }
  if (threadIdx.x == BATCH) { ws[128] = 0.f; ws[129] = 0.f; }  // 8B zero word for OOB redirect
}

__global__ __launch_bounds__(256) void bilateral_filter(
    const float* __restrict__ x, const float* __restrict__ ws, float* __restrict__ out) {
  __shared__ __align__(16) float smX[CHANS * TELEMS];  // 51.2 KB: all 32 channel tiles
  __shared__ __align__(16) float smMean[512];          // channel-mean tile (+pad)
  __shared__ __align__(16) float smDump[128];          // sink for lanes past TPAIRS

  const int tx = threadIdx.x, ty = threadIdx.y;
  const int t = ty * TILE + tx;                        // 0..255
  const int b = blockIdx.z;
  const int gx_t = blockIdx.x * TILE, gy_t = blockIdx.y * TILE;
  const int gx0 = gx_t - HALO, gy0 = gy_t - HALO;

  const float* xb = x + (size_t)b * CHANS * IMG_H * IMG_W;
  const float* zero8 = ws + 128;

  // ---- phase 1: async-DMA all 32 channel halo tiles into LDS ----
  // Each thread owns one 8-byte pair; OOB / surplus lanes are redirected to a
  // zero word (global) and a dump slot (LDS) so EXEC stays uniform: exactly
  // one async b64 per wave per channel -> 32 outstanding (<= 63 ASYNCcnt max).
  const int ly = t / 10;                 // pair row   (valid for t < 200)
  const int lx2 = (t % 10) * 2;          // pair col (even)
  const int gy = gy0 + ly, gx = gx0 + lx2;
  const bool in = (t < TPAIRS) && (gy >= 0) && (gy < IMG_H) && (gx >= 0) && (gx < IMG_W);
  const ptrdiff_t gofs = (ptrdiff_t)gy * IMG_W + gx;
  for (int c = 0; c < CHANS; ++c) {
    const void* g = in ? (const void*)(xb + (size_t)c * IMG_H * IMG_W + gofs)
                       : (const void*)zero8;
    void* l = (t < TPAIRS) ? (void*)&smX[c * TELEMS + t * 2]
                           : (void*)&smDump[(t - TPAIRS) * 2];
    async_ld_b64(l, g);
  }
  wait_async0();
  __syncthreads();

  // ---- phase 2: channel mean via chained V_WMMA_F32_16X16X4_F32 ----
  // D(16x16) = sum_{k} A(16x4) * B(4x16); A = 1/32 everywhere, B = 4 channels
  // x 16 positions, 8 chained WMMAs cover K=32. All D rows are identical, so
  // lanes 0..15 hold mean[pos] in acc[0]. EXEC is all-ones here (required).
  const int lane = t & 31, wv = t >> 5;
  v2f a; a.x = 1.f / 32.f; a.y = 1.f / 32.f;
  for (int g8 = 0; g8 < 4; ++g8) {
    const int pos16 = wv * 64 + g8 * 16;               // 8 waves x 4 groups -> 512 slots
    int p = pos16 + (lane & 15);
    p = (p < TELEMS) ? p : 0;                          // clamp pad positions (unused)
    const int cb = (lane < 16) ? 0 : 2;
    v8f acc = {0.f, 0.f, 0.f, 0.f, 0.f, 0.f, 0.f, 0.f};
#pragma unroll
    for (int kk = 0; kk < 8; ++kk) {
      v2f bv;
      bv.x = smX[(kk * 4 + cb) * TELEMS + p];
      bv.y = smX[(kk * 4 + cb + 1) * TELEMS + p];
      acc = __builtin_amdgcn_wmma_f32_16x16x4_f32(
          /*neg_a=*/false, a, /*neg_b=*/false, bv,
          /*c_mod=*/(short)0, acc, /*reuse_a=*/false, /*reuse_b=*/false);
    }
    if (lane < 16) smMean[pos16 + lane] = acc[0];
  }
  __syncthreads();

  // ---- phase 3: per-pixel combined weights ----
  // w_t = sk_t*e_t / (sum(sk*e) + 1e-8 * sum_mask(e))  == reference's
  // (sk * ck/ck.sum()) / (sum + 1e-8) normalization, epsilon included.
  const float* wp = ws + b * 32;
  const float inv2s2 = wp[25];
  const float m0 = smMean[(ty + HALO) * TW + (tx + HALO)];
  float wn[25];
  float cks = 0.f, wsm = 0.f;
#pragma unroll
  for (int dy = 0; dy < 5; ++dy)
#pragma unroll
    for (int dx = 0; dx < 5; ++dx) {
      const int tt = dy * 5 + dx;
      const float skv = wp[tt];                        // 0 exactly where masked out
      const float dv = smMean[(ty + dy) * TW + (tx + dx)] - m0;
      const float e = __expf(-(dv * dv) * inv2s2);
      if (skv > 0.f) cks += e;                         // ck.sum() over mask support
      const float w = skv * e;
      wsm += w;
      wn[tt] = w;
    }
  const float dn = 1.f / (wsm + 1e-8f * cks);
#pragma unroll
  for (int tt = 0; tt < 25; ++tt) wn[tt] *= dn;

  // ---- phase 4: 25-tap stencil for all 32 channels from resident LDS ----
  float* ob = out + (size_t)b * CHANS * IMG_H * IMG_W
                  + (size_t)(gy_t + ty) * IMG_W + (gx_t + tx);
  for (int c = 0; c < CHANS; ++c) {
    const float* row = &smX[c * TELEMS];
    float acc = 0.f;
#pragma unroll
    for (int dy = 0; dy < 5; ++dy)
#pragma unroll
      for (int dx = 0; dx < 5; ++dx)
        acc = fmaf(wn[dy * 5 + dx], row[(ty + dy) * TW + (tx + dx)], acc);
    ob[(size_t)c * IMG_H * IMG_W] = acc;
  }
}

extern "C" void kernel_launch(void* const* d_in, const int* in_sizes, int n_in,
                              void* d_out, int out_size, void* d_ws, size_t ws_size,
                              hipStream_t stream) {
  const float* x = (const float*)d_in[0];       // (4,32,256,256) f32
  const float* params = (const float*)d_in[1];  // (4,3) f32
  float* out = (float*)d_out;                   // (4,32,256,256) f32
  float* wsf = (float*)d_ws;                    // [0..127]=per-sample sk/inv2s2, [128..129]=zero

  bilateral_prep<<<dim3(1), dim3(64), 0, stream>>>(params, wsf);

  dim3 grid(IMG_W / TILE, IMG_H / TILE, BATCH);  // (16,16,4)
  dim3 block(TILE, TILE, 1);                     // 256 threads = 8 waves
  bilateral_filter<<<grid, block, 0, stream>>>(x, wsf, out);
}